// SelfAttention_627065225891
// MI455X (gfx1250) — compile-verified
//
#include <hip/hip_runtime.h>
#include <hip/hip_bf16.h>
#include <math.h>
#include <stdint.h>

// ---------------- problem constants (from reference) ----------------
constexpr int Bb   = 16;
constexpr int Ss   = 16;
constexpr int Dd   = 4096;
constexpr int Hh   = 32;
constexpr int KVH  = 8;
constexpr int HD   = Dd / Hh;        // 128
constexpr int NREP = Hh / KVH;       // 4
constexpr int MAXS = 4096;

typedef float v2f __attribute__((ext_vector_type(2)));
typedef float v8f __attribute__((ext_vector_type(8)));

// ---------------- WMMA f32 16x16x4 wrapper (probe-confirmed) ----------------
static __device__ __forceinline__ v8f wmma4(v2f a, v2f b, v8f c) {
  return __builtin_amdgcn_wmma_f32_16x16x4_f32(
      /*neg_a=*/false, a, /*neg_b=*/false, b,
      /*c_mod=*/(short)0, c, /*reuse_a=*/false, /*reuse_b=*/false);
}

// ---------------- async global -> LDS (CDNA5, ASYNCcnt) ----------------
// VDST = VGPR holding 32-bit LDS byte offset; VADDR = 64-bit global address.
static __device__ __forceinline__ void async_ld128(uint32_t lds_off, const float* g) {
  asm volatile("global_load_async_to_lds_b128 %0, %1, off"
               :: "v"(lds_off), "v"(g) : "memory");
}
static __device__ __forceinline__ uint32_t lds_off32(const void* p) {
  return (uint32_t)(uintptr_t)p;   // generic LDS pointer: low 32 bits = LDS offset
}

// =====================================================================
// Generic fp32 WMMA GEMM   C[M,N] = A[M,K] @ B[K,N]
// One wave computes a 64x16 tile (4 stacked 16x16): B-fragments are
// loaded once and reused by 4 WMMAs -> 4x better B reuse, 4x less L2
// traffic on the weight stream (Wq/Wo are 64MB, L2 is 192MB).
// A-frag: lane<16 holds A[row][k+0..1], lane>=16 holds A[row][k+2..3]
// B-frag: lane<16 holds B[k+0..1][col], lane>=16 holds B[k+2..3][col]
// C/D   : VGPR i = rows i / i+8 split across lane halves, col = lane%16
// =====================================================================
__global__ void gemm_f32_wmma(const float* __restrict__ A,
                              const float* __restrict__ B,
                              float* __restrict__ C,
                              int M, int N, int K) {
  const int wave = (blockIdx.x * blockDim.x + threadIdx.x) >> 5;
  const int lane = threadIdx.x & 31;
  const int ntiles = N >> 4;
  const int mtiles = M >> 6;                 // 64-row blocks (M % 64 == 0 here)
  if (wave >= mtiles * ntiles) return;
  const int tm = wave / ntiles;
  const int tn = wave % ntiles;
  const int half = lane >> 4;
  const int l16  = lane & 15;

  const float* ar[4];
#pragma unroll
  for (int r = 0; r < 4; ++r)
    ar[r] = A + (size_t)(tm * 64 + r * 16 + l16) * K;
  const float* bcol = B + tn * 16 + l16;

  v8f acc[4];
#pragma unroll
  for (int r = 0; r < 4; ++r) acc[r] = {};

  for (int k = 0; k < K; k += 4) {
    v2f bf;
    bf.x = bcol[(size_t)(k + half * 2)     * N];
    bf.y = bcol[(size_t)(k + half * 2 + 1) * N];
#pragma unroll
    for (int r = 0; r < 4; ++r) {
      v2f af = *(const v2f*)(ar[r] + k + half * 2);
      acc[r] = wmma4(af, bf, acc[r]);
    }
  }

#pragma unroll
  for (int r = 0; r < 4; ++r) {
#pragma unroll
    for (int i = 0; i < 8; ++i) {
      const int row = tm * 64 + r * 16 + i + half * 8;
      C[(size_t)row * N + tn * 16 + l16] = acc[r][i];
    }
  }
}

// =====================================================================
// RoPE applied in-place to xq (B*S,H,HD) and xk (B*S,KVH,HD)
// =====================================================================
__global__ void rope_kernel(float* __restrict__ xq, float* __restrict__ xk,
                            const int* __restrict__ start_pos_p) {
  const int sp = *start_pos_p;
  const int hd2 = HD / 2;
  const int qpairs = Bb * Ss * Hh * hd2;
  const int tot    = qpairs + Bb * Ss * KVH * hd2;
  int idx = blockIdx.x * blockDim.x + threadIdx.x;
  if (idx >= tot) return;

  float* ptr;
  int d, token;
  if (idx < qpairs) {
    int i = idx;
    d = i % hd2;  i /= hd2;
    int h = i % Hh; i /= Hh;        // i = token (b*S+s)
    token = i;
    ptr = xq + ((size_t)i * Hh + h) * HD + 2 * d;
  } else {
    int i = idx - qpairs;
    d = i % hd2;  i /= hd2;
    int kh = i % KVH; i /= KVH;
    token = i;
    ptr = xk + ((size_t)i * KVH + kh) * HD + 2 * d;
  }
  const int s = token % Ss;
  const float pos = (float)(sp + s);
  const float theta = __expf(-9.2103403719761836f * (2.0f * d) / (float)HD);
  const float f = pos * theta;
  const float c = __cosf(f), sn = __sinf(f);
  const float xr = ptr[0], xi = ptr[1];
  ptr[0] = xr * c - xi * sn;
  ptr[1] = xr * sn + xi * c;
}

// =====================================================================
// Flash attention. One workgroup (4 waves) per (b, kv-head): the 4 GQA
// rep-heads share K/V chunks staged in LDS via async global->LDS loads
// (double-buffered, ASYNCcnt), so the 512MB KV stream hits HBM once.
// Tokens t < start_pos come from the unmodified cache; t >= start_pos
// come from the freshly projected (functional cache update) xk/xv.
// =====================================================================
static __device__ __forceinline__ const float*
kv_row(const float* __restrict__ cache, const float* __restrict__ fresh,
       int b, int t, int kh, int sp) {
  if (t < sp) return cache + ((size_t)(b * MAXS + t) * KVH + kh) * HD;
  return fresh + ((size_t)(b * Ss + (t - sp)) * KVH + kh) * HD;
}

__global__ void attn_kernel(const float* __restrict__ xq,
                            const float* __restrict__ xk,
                            const float* __restrict__ xv,
                            const float* __restrict__ cache_k,
                            const float* __restrict__ cache_v,
                            float* __restrict__ ctx,
                            const int* __restrict__ start_pos_p) {
  __shared__ float Kbuf[2][16 * HD];      // 2 x 8KB
  __shared__ float Vbuf[2][16 * HD];      // 2 x 8KB
  __shared__ float Pbuf[NREP][16 * 16];   // per-wave P tile

  const int sp = *start_pos_p;
  const int kv_len = sp + Ss;
  const int nchunks = kv_len >> 4;

  const int wib  = threadIdx.x >> 5;      // 0..3  -> rep-head index
  const int lane = threadIdx.x & 31;
  const int half = lane >> 4;
  const int l16  = lane & 15;

  const int b  = blockIdx.x / KVH;
  const int kh = blockIdx.x % KVH;
  const int h  = kh * NREP + wib;         // this wave's query head
  float* pbuf = Pbuf[wib];

  // cooperative async staging: each wave issues 4 K rows + 4 V rows
  auto stage = [&](int sel, int t0) {
#pragma unroll
    for (int r = 0; r < 4; ++r) {
      const int tok = r * NREP + wib;
      const float* gk = kv_row(cache_k, xk, b, t0 + tok, kh, sp) + lane * 4;
      const float* gv = kv_row(cache_v, xv, b, t0 + tok, kh, sp) + lane * 4;
      async_ld128(lds_off32(&Kbuf[sel][tok * HD + lane * 4]), gk);
      async_ld128(lds_off32(&Vbuf[sel][tok * HD + lane * 4]), gv);
    }
  };

  // ---- Q tile (16 x 128) as A-fragments, pre-scaled by 1/sqrt(HD)
  const float scale = 0.08838834764831845f;   // 1/sqrt(128)
  const float* qrow = xq + ((size_t)(b * Ss + l16) * Hh + h) * HD;
  v2f qf[32];
#pragma unroll
  for (int kk = 0; kk < 32; ++kk) {
    v2f t = *(const v2f*)(qrow + kk * 4 + half * 2);
    qf[kk].x = t.x * scale;
    qf[kk].y = t.y * scale;
  }

  v8f acc[8];
#pragma unroll
  for (int j = 0; j < 8; ++j) acc[j] = {};
  float mrow[8], lrow[8];
#pragma unroll
  for (int i = 0; i < 8; ++i) { mrow[i] = -INFINITY; lrow[i] = 0.0f; }

  stage(0, 0);                             // prime the pipeline (8 async/wave)

  for (int c = 0; c < nchunks; ++c) {
    const int cur = c & 1;
    if (c + 1 < nchunks) {
      stage(cur ^ 1, (c + 1) * 16);        // 8 more in flight for next chunk
      // in-order completion: <=8 outstanding => current chunk has landed
      asm volatile("s_wait_asynccnt 0x8" ::: "memory");
    } else {
      asm volatile("s_wait_asynccnt 0x0" ::: "memory");
    }
    __syncthreads();                       // all waves' shares visible

    const float* Kc = Kbuf[cur];
    const float* Vc = Vbuf[cur];

    // ---- scores: S = Q @ K^T (16x16), K dim = 128 in steps of 4
    v8f sc = {};
#pragma unroll
    for (int kk = 0; kk < 32; ++kk) {
      v2f kb = *(const v2f*)(Kc + l16 * HD + kk * 4 + half * 2);
      sc = wmma4(qf[kk], kb, sc);
    }

    // ---- online softmax (rows live across 16-lane halves)
    float pv[8];
#pragma unroll
    for (int i = 0; i < 8; ++i) {
      float r = sc[i];
      r = fmaxf(r, __shfl_xor(r, 1));
      r = fmaxf(r, __shfl_xor(r, 2));
      r = fmaxf(r, __shfl_xor(r, 4));
      r = fmaxf(r, __shfl_xor(r, 8));
      const float mn    = fmaxf(mrow[i], r);
      const float alpha = __expf(mrow[i] - mn);
      const float p     = __expf(sc[i] - mn);
      float rs = p;
      rs += __shfl_xor(rs, 1);
      rs += __shfl_xor(rs, 2);
      rs += __shfl_xor(rs, 4);
      rs += __shfl_xor(rs, 8);
      lrow[i] = lrow[i] * alpha + rs;
      mrow[i] = mn;
      pv[i]   = p;
#pragma unroll
      for (int j = 0; j < 8; ++j) acc[j][i] *= alpha;
    }

    // ---- C-layout -> A-layout via per-wave LDS tile
#pragma unroll
    for (int i = 0; i < 8; ++i)
      pbuf[(i + half * 8) * 16 + l16] = pv[i];

    // ---- ctx += P @ V (16x16 @ 16x128): 8 N-tiles, K=16 in steps of 4
#pragma unroll
    for (int n0 = 0; n0 < 4; ++n0) {
      v2f af = *(const v2f*)(pbuf + l16 * 16 + n0 * 4 + half * 2);
      const int rA = n0 * 4 + half * 2;
#pragma unroll
      for (int j = 0; j < 8; ++j) {
        v2f bf;
        bf.x = Vc[rA * HD + j * 16 + l16];
        bf.y = Vc[(rA + 1) * HD + j * 16 + l16];
        acc[j] = wmma4(af, bf, acc[j]);
      }
    }
    __syncthreads();                       // done reading cur before reuse
  }

  // ---- normalize and store ctx in (b, s, H, HD) layout
#pragma unroll
  for (int j = 0; j < 8; ++j) {
#pragma unroll
    for (int i = 0; i < 8; ++i) {
      const int row = i + half * 8;
      ctx[((size_t)(b * Ss + row) * Hh + h) * HD + j * 16 + l16] =
          acc[j][i] / lrow[i];
    }
  }
}

// =====================================================================
// launcher
// =====================================================================
extern "C" void kernel_launch(void* const* d_in, const int* in_sizes, int n_in,
                              void* d_out, int out_size, void* d_ws, size_t ws_size,
                              hipStream_t stream) {
  (void)in_sizes; (void)n_in; (void)out_size; (void)ws_size;

  const float* x       = (const float*)d_in[0];
  const float* Wq      = (const float*)d_in[1];
  const float* Wk      = (const float*)d_in[2];
  const float* Wv      = (const float*)d_in[3];
  const float* Wo      = (const float*)d_in[4];
  const float* cache_k = (const float*)d_in[5];
  const float* cache_v = (const float*)d_in[6];
  const int*   sp      = (const int*)d_in[7];
  float* out = (float*)d_out;

  float* ws  = (float*)d_ws;
  float* xq  = ws;                                        // 256*4096
  float* xk  = xq + (size_t)Bb * Ss * Hh  * HD;           // 256*1024
  float* xv  = xk + (size_t)Bb * Ss * KVH * HD;           // 256*1024
  float* ctx = xv + (size_t)Bb * Ss * KVH * HD;           // 256*4096

  const dim3 blk(256);                                    // 8 waves / block
  auto blocks_for = [](int M, int N) {
    int tiles = (M / 64) * (N / 16);
    return (tiles + 7) / 8;
  };

  const int M = Bb * Ss;   // 256

  // QKV projections
  gemm_f32_wmma<<<blocks_for(M, Hh * HD),  blk, 0, stream>>>(x, Wq, xq, M, Hh * HD,  Dd);
  gemm_f32_wmma<<<blocks_for(M, KVH * HD), blk, 0, stream>>>(x, Wk, xk, M, KVH * HD, Dd);
  gemm_f32_wmma<<<blocks_for(M, KVH * HD), blk, 0, stream>>>(x, Wv, xv, M, KVH * HD, Dd);

  // RoPE on q, k
  const int npairs = Bb * Ss * (Hh + KVH) * (HD / 2);
  rope_kernel<<<(npairs + 255) / 256, blk, 0, stream>>>(xq, xk, sp);

  // attention: one 4-wave workgroup per (b, kv-head); rep-heads share K/V
  attn_kernel<<<Bb * KVH, dim3(32 * NREP), 0, stream>>>(xq, xk, xv,
                                                        cache_k, cache_v, ctx, sp);

  // output projection
  gemm_f32_wmma<<<blocks_for(M, Dd), blk, 0, stream>>>(ctx, Wo, out, M, Dd, Hh * HD);
}